// BaseNQS_6021544149398
// MI455X (gfx1250) — compile-verified
//
#include <hip/hip_runtime.h>

typedef float v2f __attribute__((ext_vector_type(2)));
typedef float v8f __attribute__((ext_vector_type(8)));

#define NSITE 100
#define DDIM  128
#define NOCC  50
#define LDP   52   // phi leading dim (pad to dodge bank conflicts)

__launch_bounds__(256, 1)
__global__ void nqs_psi_kernel(const int*   __restrict__ configs,
                               const float* __restrict__ tok,
                               const float* __restrict__ pos,
                               const float* __restrict__ W,
                               const float* __restrict__ bias,
                               float*       __restrict__ out)
{
    __shared__ float Hs[2][64 * DDIM];        // gathered h rows, zero padded to 64
    __shared__ float Phi[8][NOCC * LDP];      // one 50x50 matrix per wave
    __shared__ int   Idx[2][64];
    __shared__ float Dets[8];

    const int b   = blockIdx.x;
    const int tid = threadIdx.x;
    const int* cfg = configs + b * NSITE;

    // ---- occupied-index build (thread 0: up, thread 1: dn) ----
    // Matches stable argsort(-mask)[:50] then sort: 50 smallest occupied
    // indices; if fewer than 50 occupied, fill with smallest unoccupied;
    // compaction in ascending site order is already sorted.
    if (tid < 2) {
        const int spin = tid;
        int cntOcc = 0;
        for (int s = 0; s < NSITE; s++) {
            int c = cfg[s];
            bool o = (spin == 0) ? ((c == 1) || (c == 3)) : ((c == 2) || (c == 3));
            cntOcc += o ? 1 : 0;
        }
        int capped = (cntOcc < NOCC) ? cntOcc : NOCC;
        int need = NOCC - capped;
        int taken = 0, fill = 0, p = 0;
        for (int s = 0; s < NSITE; s++) {
            int c = cfg[s];
            bool o = (spin == 0) ? ((c == 1) || (c == 3)) : ((c == 2) || (c == 3));
            bool sel = false;
            if (o) { if (taken < NOCC) sel = true; taken++; }
            else if (fill < need) { sel = true; fill++; }
            if (sel) Idx[spin][p++] = s;
        }
        for (int i = p; i < 64; i++) Idx[spin][i] = -1;
    }
    __syncthreads();

    // ---- stage H = tok_table[cfg[site]] + pos_table[site] for gathered rows ----
    for (int e = tid; e < 2 * 64 * DDIM; e += 256) {
        int spin = e / (64 * DDIM);
        int rem  = e - spin * 64 * DDIM;
        int i = rem >> 7;
        int d = rem & (DDIM - 1);
        int site = Idx[spin][i];
        float v = 0.0f;
        if (site >= 0) {
            int c = cfg[site];
            v = tok[c * DDIM + d] + pos[site * DDIM + d];
        }
        Hs[spin][i * DDIM + d] = v;
    }
    __syncthreads();

    // ---- WMMA GEMM: each wave computes its 50x50 phi = H_occ @ W_slice^T + b ----
    // Loop order: nt outer; per k-step load ONE B fragment, feed FOUR M-tile
    // accumulators (4x v8f = 32 VGPRs). Keeps live VGPRs < 256 (no MSB thrash)
    // and cuts global W fragment traffic 4x.
    const int w    = tid >> 5;         // wave id 0..7
    const int lane = tid & 31;
    const int spin = w >> 2;           // 0 = up, 1 = dn
    const int kdet = w & 3;            // determinant index k
    const int wrowBase = spin * (4 * NOCC) + kdet * NOCC;   // W_orb row offset
    const int lm = lane & 15;          // row/col within 16-tile
    const int kh = lane >> 4;          // 0: K=0,1  1: K=2,3  (per ISA 16x4 layout)

    const float* Hrow = &Hs[spin][0];
    float* Mw = &Phi[w][0];

    const float* A0 = Hrow + (0 * 16 + lm) * DDIM + kh * 2;
    const float* A1 = A0 + 16 * DDIM;
    const float* A2 = A1 + 16 * DDIM;
    const float* A3 = A2 + 16 * DDIM;

    #pragma unroll 1
    for (int nt = 0; nt < 4; nt++) {
        const int col = nt * 16 + lm;
        // Columns >= 50 compute garbage that is never stored (guard below);
        // clamp the pointer instead of masking every load.
        const int colc = (col < NOCC) ? col : (NOCC - 1);
        const float* Bp = W + (wrowBase + colc) * DDIM + kh * 2;

        v8f acc0 = {0.f,0.f,0.f,0.f,0.f,0.f,0.f,0.f};
        v8f acc1 = acc0, acc2 = acc0, acc3 = acc0;

        #pragma unroll 4
        for (int kk = 0; kk < DDIM; kk += 4) {
            v2f bb; bb.x = Bp[kk]; bb.y = Bp[kk + 1];
            v2f a0; a0.x = A0[kk]; a0.y = A0[kk + 1];
            v2f a1; a1.x = A1[kk]; a1.y = A1[kk + 1];
            v2f a2; a2.x = A2[kk]; a2.y = A2[kk + 1];
            v2f a3; a3.x = A3[kk]; a3.y = A3[kk + 1];
            acc0 = __builtin_amdgcn_wmma_f32_16x16x4_f32(false, a0, false, bb, (short)0, acc0, false, false);
            acc1 = __builtin_amdgcn_wmma_f32_16x16x4_f32(false, a1, false, bb, (short)0, acc1, false, false);
            acc2 = __builtin_amdgcn_wmma_f32_16x16x4_f32(false, a2, false, bb, (short)0, acc2, false, false);
            acc3 = __builtin_amdgcn_wmma_f32_16x16x4_f32(false, a3, false, bb, (short)0, acc3, false, false);
        }

        // C layout: VGPR r, lanes 0-15 -> M = mt*16 + r, lanes 16-31 -> M = mt*16 + 8 + r
        const int cn = col;
        const float bj = bias[wrowBase + colc];
        if (cn < NOCC) {
            #pragma unroll
            for (int r = 0; r < 8; r++) {
                int cm0 = 0 * 16 + kh * 8 + r;
                int cm1 = 1 * 16 + kh * 8 + r;
                int cm2 = 2 * 16 + kh * 8 + r;
                int cm3 = 3 * 16 + kh * 8 + r;
                Mw[cm0 * LDP + cn] = acc0[r] + bj;
                Mw[cm1 * LDP + cn] = acc1[r] + bj;
                Mw[cm2 * LDP + cn] = acc2[r] + bj;
                if (cm3 < NOCC) Mw[cm3 * LDP + cn] = acc3[r] + bj;
            }
        }
    }
    __syncthreads();

    // ---- per-wave LU determinant with partial pivoting (in LDS) ----
    float detv = 1.0f;
    for (int c = 0; c < NOCC; c++) {
        // lane-parallel pivot search over rows c..49
        float best = -1.0f; int bestr = c;
        for (int r = c + lane; r < NOCC; r += 32) {
            float v = fabsf(Mw[r * LDP + c]);
            if (v > best) { best = v; bestr = r; }
        }
        for (int off = 16; off > 0; off >>= 1) {
            float ob = __shfl_down(best, off);
            int   orr = __shfl_down(bestr, off);
            if (ob > best) { best = ob; bestr = orr; }
        }
        bestr = __shfl(bestr, 0);
        if (bestr != c) {
            for (int j = lane; j < NOCC; j += 32) {
                float t = Mw[c * LDP + j];
                Mw[c * LDP + j] = Mw[bestr * LDP + j];
                Mw[bestr * LDP + j] = t;
            }
            detv = -detv;
        }
        __syncthreads();
        float piv = Mw[c * LDP + c];
        detv *= piv;
        float inv = 1.0f / piv;
        for (int r = c + 1 + lane; r < NOCC; r += 32) {
            float f = Mw[r * LDP + c] * inv;
            for (int j = c; j < NOCC; j++)
                Mw[r * LDP + j] -= f * Mw[c * LDP + j];
        }
        __syncthreads();
    }
    if (lane == 0) Dets[w] = detv;
    __syncthreads();

    // ---- combine: psi = sum_k det_up[k] * det_dn[k] ----
    if (tid == 0) {
        float psi = 0.0f;
        for (int k = 0; k < 4; k++) psi += Dets[k] * Dets[4 + k];
        out[2 * b + 0] = logf(fabsf(psi) + 1e-30f);
        out[2 * b + 1] = (psi >= 0.0f) ? 0.0f : 3.14159265358979f;
    }
}

extern "C" void kernel_launch(void* const* d_in, const int* in_sizes, int n_in,
                              void* d_out, int out_size, void* d_ws, size_t ws_size,
                              hipStream_t stream) {
    const int*   configs = (const int*)d_in[0];    // (B, 100) int
    const float* tok     = (const float*)d_in[1];  // (4, 128)
    const float* pos     = (const float*)d_in[2];  // (144, 128)
    const float* W       = (const float*)d_in[3];  // (400, 128)
    const float* bias    = (const float*)d_in[4];  // (400,)
    float* out = (float*)d_out;                    // (B,) complex64 = 2B floats

    int B = in_sizes[0] / NSITE;
    nqs_psi_kernel<<<B, 256, 0, stream>>>(configs, tok, pos, W, bias, out);
}